// Decoder_46170898432436
// MI455X (gfx1250) — compile-verified
//
#include <hip/hip_runtime.h>
#include <hip/hip_bf16.h>
#include <math.h>
#include <stdint.h>

// Sizes from the reference
#define S_  128
#define B_  64
#define H_  1024
#define E_  512
#define V_  32000

typedef __attribute__((ext_vector_type(2))) float v2f;
typedef __attribute__((ext_vector_type(8))) float v8f;

// ---------------------------------------------------------------------------
// Generic f32 WMMA GEMM:  C[M,N] = (acc? C : 0) + A[M,K] @ B[N,K]^T + bias[N]
// Block tile 32(M) x 64(N), 8 waves (2x4 of 16x16 WMMA tiles), K-step 16.
// Tiles are staged into LDS with CDNA5 async global->LDS copies (ASYNCcnt),
// bypassing VGPRs entirely; one s_wait_asynccnt 0 before the block barrier.
// Requires M%32==0, N%64==0, K%16==0 (true for all uses here).
// grid = dim3(N/64, M/32), block = 256
// ---------------------------------------------------------------------------
__global__ __launch_bounds__(256)
void gemm_wmma_f32(const float* __restrict__ A, int lda,
                   const float* __restrict__ Bm, int ldb,
                   float* __restrict__ C, int ldc,
                   int K, const float* __restrict__ bias, int accflag)
{
    __shared__ alignas(16) float As[32][20];
    __shared__ alignas(16) float Bs[64][20];
    const int m0 = blockIdx.y * 32;
    const int n0 = blockIdx.x * 64;
    const int tid  = threadIdx.x;
    const int wave = tid >> 5;
    const int lane = tid & 31;
    const int wm = wave >> 2;        // 0..1
    const int wn = wave & 3;         // 0..3
    const int row = lane & 15;
    const int kk2 = (lane >> 4) * 2; // K sub-offset per ISA A/B f32 layout

    // Per-thread staging assignments (fixed across K loop)
    const int ra = tid >> 3;         // A tile row 0..31
    const int ca = (tid & 7) * 2;    // A tile col pair (8B aligned)
    const int rb = tid >> 2;         // B tile row 0..63
    const int cb = (tid & 3) * 4;    // B tile col quad (16B aligned)
    const unsigned lds_a = (unsigned)(size_t)&As[ra][ca];
    const unsigned lds_b = (unsigned)(size_t)&Bs[rb][cb];
    const float* gA = A  + (size_t)(m0 + ra) * lda + ca;
    const float* gB = Bm + (size_t)(n0 + rb) * ldb + cb;

    v8f acc = {};
    for (int k0 = 0; k0 < K; k0 += 16) {
        // Async global->LDS staging: A tile 32x16 (b64/lane), B tile 64x16 (b128/lane)
        asm volatile("global_load_async_to_lds_b64 %0, %1, off"
                     :: "v"(lds_a),
                        "v"((unsigned long long)(size_t)(gA + k0))
                     : "memory");
        asm volatile("global_load_async_to_lds_b128 %0, %1, off"
                     :: "v"(lds_b),
                        "v"((unsigned long long)(size_t)(gB + k0))
                     : "memory");
        if (k0 + 16 < K)
            __builtin_prefetch(gB + k0 + 16, 0, 0);   // global_prefetch_b8
        asm volatile("s_wait_asynccnt 0x0" ::: "memory");
        __syncthreads();
#pragma unroll
        for (int kk = 0; kk < 16; kk += 4) {
            v2f a, b;
            a.x = As[wm * 16 + row][kk + kk2];
            a.y = As[wm * 16 + row][kk + kk2 + 1];
            b.x = Bs[wn * 16 + row][kk + kk2];
            b.y = Bs[wn * 16 + row][kk + kk2 + 1];
            acc = __builtin_amdgcn_wmma_f32_16x16x4_f32(
                false, a, false, b, (short)0, acc, false, false);
        }
        __syncthreads();
    }

    const int n = n0 + wn * 16 + (lane & 15);
    const float bv = bias ? bias[n] : 0.0f;
#pragma unroll
    for (int i = 0; i < 8; i++) {
        int m = m0 + wm * 16 + i + 8 * (lane >> 4);
        float val = acc[i] + bv;
        if (accflag) val += C[m * ldc + n];
        C[m * ldc + n] = val;
    }
}

// ---------------------------------------------------------------------------
// Fused attention logits: for M = S*B rows of enc,
//   logits[m] = v . tanh(u[b(m), :] + enc[m,:] @ Wenc^T) + v_b - 1e5*pad(m)
// Wenc[h,k] = W_a_w[h, H + k].  One block per 16-row M tile; 8 waves each
// cover 8 of the 64 h-tiles (full K=1024 WMMA loop), epilogue does
// tanh * v[h] and reduces across the N dimension with LDS float atomics.
// enc/Wenc stream through L2 (both fit easily in the 192 MB L2).
// ---------------------------------------------------------------------------
__global__ __launch_bounds__(256)
void attn_logits_kernel(const float* __restrict__ enc,   // [S*B, H]
                        const float* __restrict__ Waw,   // [H, 2H]
                        const float* __restrict__ u,     // [B, H]
                        const float* __restrict__ vw,    // [H]
                        const float* __restrict__ vb,    // [1]
                        float* __restrict__ att_logits)  // [S*B]
{
    __shared__ float red[16];
    __shared__ float padf[16];
    const int m0  = blockIdx.x * 16;
    const int tid = threadIdx.x;

    if (tid < 16) {
        red[tid] = 0.0f;
        const float* ar = enc + (size_t)(m0 + tid) * H_;
        float s = 0.0f;
        for (int c = 0; c < H_; ++c) s += ar[c];
        padf[tid] = (s == 0.0f) ? 1.0f : 0.0f;
    }
    __syncthreads();

    const int wave = tid >> 5;
    const int lane = tid & 31;
    const int row  = lane & 15;
    const int kk2  = (lane >> 4) * 2;
    const float* Arow = enc + (size_t)(m0 + row) * H_;

    for (int ht = wave; ht < H_ / 16; ht += 8) {
        const int h = ht * 16 + (lane & 15);
        const float* Brow = Waw + (size_t)h * (2 * H_) + H_;  // Wenc[h, :]
        v8f acc = {};
        for (int k = 0; k < H_; k += 4) {
            v2f a, b;
            a.x = Arow[k + kk2];
            a.y = Arow[k + kk2 + 1];
            b.x = Brow[k + kk2];
            b.y = Brow[k + kk2 + 1];
            acc = __builtin_amdgcn_wmma_f32_16x16x4_f32(
                false, a, false, b, (short)0, acc, false, false);
        }
        const float vh = vw[h];
#pragma unroll
        for (int i = 0; i < 8; i++) {
            int ml = i + 8 * (lane >> 4);
            int bb = (m0 + ml) & (B_ - 1);
            float e = tanhf(acc[i] + u[bb * H_ + h]);
            atomicAdd(&red[ml], vh * e);   // ds_add_f32
        }
    }
    __syncthreads();
    if (tid < 16)
        att_logits[m0 + tid] = red[tid] + vb[0] - 100000.0f * padf[tid];
}

// Softmax over encoder time S per batch column b. block = S threads.
__global__ void attn_softmax_kernel(const float* __restrict__ lg,
                                    float* __restrict__ attn)
{
    __shared__ float sred[S_];
    const int b = blockIdx.x;
    const int s = threadIdx.x;
    float val = lg[s * B_ + b];
    sred[s] = val; __syncthreads();
    for (int k = S_ / 2; k > 0; k >>= 1) {
        if (s < k) sred[s] = fmaxf(sred[s], sred[s + k]);
        __syncthreads();
    }
    float mx = sred[0]; __syncthreads();
    float e = __expf(val - mx);
    sred[s] = e; __syncthreads();
    for (int k = S_ / 2; k > 0; k >>= 1) {
        if (s < k) sred[s] += sred[s + k];
        __syncthreads();
    }
    attn[s * B_ + b] = e / sred[0];
}

// context[b,h] = sum_s attn[s,b] * enc[s,b,h]   -> writes x[b, 0:H]
__global__ __launch_bounds__(256)
void context_kernel(const float* __restrict__ attn,
                    const float* __restrict__ enc,
                    float* __restrict__ x)  // [B, H+E]
{
    __shared__ float aw[S_];
    const int b = blockIdx.x;
    const int tid = threadIdx.x;
    if (tid < S_) aw[tid] = attn[tid * B_ + b];
    __syncthreads();
    for (int h = tid; h < H_; h += 256) {
        float s = 0.0f;
        for (int t = 0; t < S_; ++t)
            s += aw[t] * enc[((size_t)t * B_ + b) * H_ + h];
        x[b * (H_ + E_) + h] = s;
    }
}

// embedded = embedding[last_word]  -> writes x[b, H:H+E]
__global__ void embed_kernel(const int* __restrict__ last_word,
                             const float* __restrict__ emb,
                             float* __restrict__ x)
{
    int idx = blockIdx.x * 256 + threadIdx.x;   // B*E
    if (idx >= B_ * E_) return;
    int b = idx >> 9, e = idx & (E_ - 1);
    x[b * (H_ + E_) + H_ + e] = emb[(size_t)last_word[b] * E_ + e];
}

// LSTM cell elementwise (PyTorch gate order i,f,g,o)
__global__ void lstm_kernel(const float* __restrict__ gates,  // [B, 4H]
                            const float* __restrict__ c_old,  // [B, H]
                            float* __restrict__ h_out,
                            float* __restrict__ c_out)
{
    int idx = blockIdx.x * 256 + threadIdx.x;   // B*H
    int b = idx >> 10, n = idx & (H_ - 1);
    const float* g = gates + (size_t)b * 4 * H_;
    float ig = 1.0f / (1.0f + __expf(-g[n]));
    float fg = 1.0f / (1.0f + __expf(-g[H_ + n]));
    float gg = tanhf(g[2 * H_ + n]);
    float og = 1.0f / (1.0f + __expf(-g[3 * H_ + n]));
    float c = fg * c_old[idx] + ig * gg;
    c_out[idx] = c;
    h_out[idx] = og * tanhf(c);
}

// In-place log-softmax over V per row. One block per b.
__global__ __launch_bounds__(256)
void logsoftmax_kernel(float* __restrict__ logits)
{
    __shared__ float sred[256];
    const int b = blockIdx.x;
    float* row = logits + (size_t)b * V_;
    const int tid = threadIdx.x;

    float mx = -3.0e38f;
    for (int v = tid; v < V_; v += 256) mx = fmaxf(mx, row[v]);
    sred[tid] = mx; __syncthreads();
    for (int k = 128; k > 0; k >>= 1) {
        if (tid < k) sred[tid] = fmaxf(sred[tid], sred[tid + k]);
        __syncthreads();
    }
    mx = sred[0]; __syncthreads();

    float sum = 0.0f;
    for (int v = tid; v < V_; v += 256) sum += __expf(row[v] - mx);
    sred[tid] = sum; __syncthreads();
    for (int k = 128; k > 0; k >>= 1) {
        if (tid < k) sred[tid] += sred[tid + k];
        __syncthreads();
    }
    const float lse = mx + __logf(sred[0]);
    __syncthreads();

    for (int v = tid; v < V_; v += 256) row[v] = row[v] - lse;
}

// ---------------------------------------------------------------------------
extern "C" void kernel_launch(void* const* d_in, const int* in_sizes, int n_in,
                              void* d_out, int out_size, void* d_ws, size_t ws_size,
                              hipStream_t stream)
{
    const float* hidden_h  = (const float*)d_in[0];
    const float* hidden_c  = (const float*)d_in[1];
    const float* enc       = (const float*)d_in[2];
    const int*   last_word = (const int*)  d_in[3];
    const float* embedding = (const float*)d_in[4];
    const float* W_a_w     = (const float*)d_in[5];
    const float* W_a_b     = (const float*)d_in[6];
    const float* v_w       = (const float*)d_in[7];
    const float* v_b       = (const float*)d_in[8];
    const float* W_ih      = (const float*)d_in[9];
    const float* W_hh      = (const float*)d_in[10];
    const float* b_ih      = (const float*)d_in[11];
    const float* b_hh      = (const float*)d_in[12];
    const float* out_w     = (const float*)d_in[13];
    const float* out_b     = (const float*)d_in[14];

    // Workspace layout (floats): ~1.7 MB total
    float* ws      = (float*)d_ws;
    float* u       = ws;                              // B*H      = 65536
    float* x       = u + B_ * H_;                     // B*(H+E)  = 98304
    float* att_lg  = x + B_ * (H_ + E_);              // S*B      = 8192
    float* attn    = att_lg + S_ * B_;                // S*B      = 8192
    float* gates   = attn + S_ * B_;                  // B*4H     = 262144

    // Output layout: logp [B,V], h [B,H], c [B,H]
    float* logp  = (float*)d_out;
    float* h_out = logp + (size_t)B_ * V_;
    float* c_out = h_out + B_ * H_;

    // 1) embedding gather into x[:, H:]
    embed_kernel<<<(B_ * E_ + 255) / 256, 256, 0, stream>>>(last_word, embedding, x);

    // 2) u = hidden_h @ W_a[:, :H]^T + W_a_b   (M=64, N=1024, K=1024)
    gemm_wmma_f32<<<dim3(H_ / 64, B_ / 32), 256, 0, stream>>>(
        hidden_h, H_, W_a_w, 2 * H_, u, H_, H_, W_a_b, 0);

    // 3) fused attention energy + v-dot + pad mask  (M = S*B tiles of 16)
    attn_logits_kernel<<<(S_ * B_) / 16, 256, 0, stream>>>(
        enc, W_a_w, u, v_w, v_b, att_lg);

    // 4) softmax over S
    attn_softmax_kernel<<<B_, S_, 0, stream>>>(att_lg, attn);

    // 5) context into x[:, :H]
    context_kernel<<<B_, 256, 0, stream>>>(attn, enc, x);

    // 6) gates = x @ W_ih^T + b_ih            (M=64, N=4096, K=1536)
    gemm_wmma_f32<<<dim3(4 * H_ / 64, B_ / 32), 256, 0, stream>>>(
        x, H_ + E_, W_ih, H_ + E_, gates, 4 * H_, H_ + E_, b_ih, 0);

    // 7) gates += hidden_h @ W_hh^T + b_hh    (M=64, N=4096, K=1024)
    gemm_wmma_f32<<<dim3(4 * H_ / 64, B_ / 32), 256, 0, stream>>>(
        hidden_h, H_, W_hh, H_, gates, 4 * H_, H_, b_hh, 1);

    // 8) LSTM cell -> h_out, c_out (d_out regions)
    lstm_kernel<<<(B_ * H_) / 256, 256, 0, stream>>>(gates, hidden_c, h_out, c_out);

    // 9) vocab logits straight into d_out     (M=64, N=32000, K=1024)
    gemm_wmma_f32<<<dim3(V_ / 64, B_ / 32), 256, 0, stream>>>(
        h_out, H_, out_w, H_, logp, V_, H_, out_b, 0);

    // 10) in-place log-softmax over V
    logsoftmax_kernel<<<B_, 256, 0, stream>>>(logp);
}